// DiracGraphConv_85736137163288
// MI455X (gfx1250) — compile-verified
//
#include <hip/hip_runtime.h>

#define NN 100000
#define NE 1600000
#define D 64
#define EPS_NORM 1e-9f
#define EPS_DENOM 1e-9f

typedef float v2f __attribute__((ext_vector_type(2)));
typedef float v8f __attribute__((ext_vector_type(8)));

// Monotonic order-preserving encoding of float into uint for atomicMax.
__device__ __forceinline__ unsigned fenc(float f) {
    unsigned u = __float_as_uint(f);
    return (u & 0x80000000u) ? ~u : (u | 0x80000000u);
}
__device__ __forceinline__ float fdec(unsigned u) {
    u = (u & 0x80000000u) ? (u & 0x7FFFFFFFu) : ~u;
    return __uint_as_float(u);
}

__global__ void init_max_kernel(unsigned* maxenc) {
    *maxenc = 0x007FFFFFu;  // fenc(-inf)
}

// Per-node L2 norm of z rows.
__global__ void norm_kernel(const float* __restrict__ z, float* __restrict__ nrm) {
    int n = blockIdx.x * blockDim.x + threadIdx.x;
    if (n >= NN) return;
    const float4* zr = (const float4*)(z + (size_t)n * D);
    float s = 0.f;
#pragma unroll
    for (int i = 0; i < D / 4; ++i) {
        float4 a = zr[i];
        s += a.x * a.x + a.y * a.y + a.z * a.z + a.w * a.w;
    }
    nrm[n] = sqrtf(s);
}

// Per-edge cosine logit + global max reduction.
__global__ void logit_kernel(const float* __restrict__ z, const int* __restrict__ ei,
                             const float* __restrict__ nrm,
                             const float* __restrict__ alpha_p,
                             const float* __restrict__ bias_p,
                             float* __restrict__ logits, unsigned* __restrict__ maxenc) {
    int e = blockIdx.x * blockDim.x + threadIdx.x;
    float lg = -__builtin_inff();
    if (e < NE) {
        int r = ei[e];
        int c = ei[NE + e];
        const float4* zi = (const float4*)(z + (size_t)r * D);
        const float4* zj = (const float4*)(z + (size_t)c * D);
        float num = 0.f;
#pragma unroll
        for (int i = 0; i < D / 4; ++i) {
            float4 a = zi[i], b = zj[i];
            num += a.x * b.x + a.y * b.y + a.z * b.z + a.w * b.w;
        }
        float den = fmaxf(nrm[r] * nrm[c], EPS_NORM);
        lg = alpha_p[0] * (num / den) + bias_p[0];
        logits[e] = lg;
    }
    __shared__ float smax[256];
    smax[threadIdx.x] = lg;
    __syncthreads();
#pragma unroll
    for (int s = 128; s > 0; s >>= 1) {
        if (threadIdx.x < s) smax[threadIdx.x] = fmaxf(smax[threadIdx.x], smax[threadIdx.x + s]);
        __syncthreads();
    }
    if (threadIdx.x == 0) atomicMax(maxenc, fenc(smax[0]));
}

// exp(logit - max) in place, segment-sum denominators by row.
__global__ void exp_denom_kernel(const int* __restrict__ ei,
                                 const unsigned* __restrict__ maxenc,
                                 float* __restrict__ logits, float* __restrict__ denom) {
    int e = blockIdx.x * blockDim.x + threadIdx.x;
    if (e >= NE) return;
    float m = fdec(*maxenc);
    float w = expf(logits[e] - m);
    logits[e] = w;  // now holds exp weight
    atomicAdd(&denom[ei[e]], w);
}

// One thread per (edge, feature): coalesced gather of x[col] row, coalesced
// scatter-add into agg[row] row. All L2-resident.
__global__ void scatter_kernel(const float* __restrict__ x, const int* __restrict__ ei,
                               const float* __restrict__ expw,
                               const float* __restrict__ denom, float* __restrict__ agg) {
    int gid = blockIdx.x * blockDim.x + threadIdx.x;  // < NE*64 = 102.4M
    int e = gid >> 6;
    int d = gid & 63;
    if (e >= NE) return;
    int r = ei[e];
    int c = ei[NE + e];
    float attn = expw[e] / (denom[r] + EPS_DENOM);
    atomicAdd(&agg[r * D + d], attn * x[c * D + d]);
}

// out = agg @ W^T + b using V_WMMA_F32_16X16X4_F32.
// One wave per 16x16 output tile; K=64 -> 16 chained WMMAs.
// A-fragment (16x4, MxK): lanes 0-15 M=lane K={k,k+1}; lanes 16-31 M=lane-16 K={k+2,k+3}.
// B-fragment (4x16, KxN): lanes 0-15 N=lane K={k,k+1}; lanes 16-31 N=lane-16 K={k+2,k+3}.
// B[k][n] = W[n][k]  (row n of W supplies column n of B).
__global__ void gemm_wmma_kernel(const float* __restrict__ agg, const float* __restrict__ W,
                                 const float* __restrict__ bias, float* __restrict__ out) {
    int wave = (blockIdx.x * blockDim.x + threadIdx.x) >> 5;
    int lane = threadIdx.x & 31;
    int mt = wave >> 2;       // 100000/16 = 6250 m-tiles
    int nt = wave & 3;        // 64/16   = 4    n-tiles
    int m0 = mt * 16;
    int n0 = nt * 16;
    int half = lane >> 4;     // 0: K pair {k,k+1}; 1: K pair {k+2,k+3}
    int l = lane & 15;

    const float* arow = agg + (m0 + l) * D;  // A row M = l (per half-wave layout)
    const float* brow = W + (n0 + l) * D;    // W row n supplies B column n

    v8f c = {};
#pragma unroll
    for (int k = 0; k < D; k += 4) {
        int kk = k + 2 * half;
        v2f a = *(const v2f*)(arow + kk);
        v2f b = *(const v2f*)(brow + kk);
        c = __builtin_amdgcn_wmma_f32_16x16x4_f32(
            /*neg_a=*/false, a, /*neg_b=*/false, b,
            /*c_mod=*/(short)0, c, /*reuse_a=*/false, /*reuse_b=*/false);
    }

    float bv = bias[n0 + l];
    // C/D layout: VGPR r -> lanes 0-15: M=r, N=l ; lanes 16-31: M=r+8, N=l.
#pragma unroll
    for (int r = 0; r < 8; ++r) {
        out[(m0 + r + 8 * half) * D + n0 + l] = c[r] + bv;
    }
}

extern "C" void kernel_launch(void* const* d_in, const int* in_sizes, int n_in,
                              void* d_out, int out_size, void* d_ws, size_t ws_size,
                              hipStream_t stream) {
    const float* x         = (const float*)d_in[0];
    const int*   ei        = (const int*)d_in[1];   // [2, NE]: rows then cols
    const float* z         = (const float*)d_in[2];
    const float* W         = (const float*)d_in[3];
    const float* b         = (const float*)d_in[4];
    const float* alpha     = (const float*)d_in[5];
    const float* bias_edge = (const float*)d_in[6];
    float* out = (float*)d_out;

    // Workspace layout (floats): nrm[NN] | logits/expw[NE] | denom[NN] | agg[NN*D] | maxenc
    float* nrm    = (float*)d_ws;
    float* logits = nrm + NN;
    float* denom  = logits + NE;
    float* agg    = denom + NN;
    unsigned* maxenc = (unsigned*)(agg + (size_t)NN * D);

    // denom + agg are contiguous -> one async memset (graph-capturable).
    hipMemsetAsync(denom, 0, sizeof(float) * ((size_t)NN + (size_t)NN * D), stream);
    init_max_kernel<<<1, 1, 0, stream>>>(maxenc);

    norm_kernel<<<(NN + 255) / 256, 256, 0, stream>>>(z, nrm);
    logit_kernel<<<NE / 256, 256, 0, stream>>>(z, ei, nrm, alpha, bias_edge, logits, maxenc);
    exp_denom_kernel<<<NE / 256, 256, 0, stream>>>(ei, maxenc, logits, denom);
    scatter_kernel<<<(NE * 64) / 256, 256, 0, stream>>>(x, ei, logits, denom, agg);

    // 6250 m-tiles * 4 n-tiles = 25000 waves; 8 waves (256 threads) per block.
    gemm_wmma_kernel<<<(NN / 16 * 4) / 8, 256, 0, stream>>>(agg, W, b, out);
}